// SomaODE_65352222376339
// MI455X (gfx1250) — compile-verified
//
#include <hip/hip_runtime.h>
#include <math.h>

// LIF first-spike time, closed form:
//   a = 1 - dt/tau = 0.8
//   n = max(ceil(log1p(-1/I)/log(a)), 1)
//   out = (I > 1 && n <= MAX_STEPS) ? n*dt : +inf
//
// Memory-bound elementwise stream (256 MB in + 256 MB out, each touched once):
//  - 128-bit vector loads/stores (global_load_b128/global_store_b128)
//  - non-temporal TH hints so the >L2-sized stream doesn't thrash the 192 MB L2
//  - log1p(-1/I) = log2(I-1) - log2(I), folded scale -> two v_log_f32 per elem

#define LIF_DT        0.01f
#define LIF_TAU       0.05f
#define LIF_MAXSTEPS  200000.0f

typedef float v4f __attribute__((ext_vector_type(4)));

__device__ __forceinline__ float lif_elem(float I, float inv_log2_a) {
    // n_real = (log2(I-1) - log2(I)) * (1/log2(0.8))
    // For I <= 1: log2(I-1) is NaN/-inf -> masked out by (I > 1) predicate.
    float nr = (__log2f(I - 1.0f) - __log2f(I)) * inv_log2_a;
    float n  = fmaxf(__builtin_ceilf(nr), 1.0f);   // fmax also absorbs NaN -> 1.0
    bool valid = (I > 1.0f) && (n <= LIF_MAXSTEPS);
    return valid ? n * LIF_DT : __builtin_inff();
}

__global__ __launch_bounds__(256) void lif_v4_kernel(const v4f* __restrict__ in,
                                                     v4f* __restrict__ out,
                                                     int n4, float inv_log2_a) {
    int idx = blockIdx.x * blockDim.x + threadIdx.x;
    if (idx >= n4) return;
    // Single-use stream: non-temporal 128-bit load (th:TH_LOAD_NT)
    v4f iv = __builtin_nontemporal_load(in + idx);
    v4f r;
#pragma unroll
    for (int k = 0; k < 4; ++k) {
        r[k] = lif_elem(iv[k], inv_log2_a);
    }
    // Output never re-read on device: non-temporal 128-bit store (th:TH_STORE_NT)
    __builtin_nontemporal_store(r, out + idx);
}

__global__ __launch_bounds__(256) void lif_tail_kernel(const float* __restrict__ in,
                                                       float* __restrict__ out,
                                                       int start, int total,
                                                       float inv_log2_a) {
    int idx = start + blockIdx.x * blockDim.x + threadIdx.x;
    if (idx >= total) return;
    float I = __builtin_nontemporal_load(in + idx);
    __builtin_nontemporal_store(lif_elem(I, inv_log2_a), out + idx);
}

extern "C" void kernel_launch(void* const* d_in, const int* in_sizes, int n_in,
                              void* d_out, int out_size, void* d_ws, size_t ws_size,
                              hipStream_t stream) {
    const float* I  = (const float*)d_in[0];
    float*       out = (float*)d_out;
    const int total = in_sizes[0];              // 64 * 1,000,000 = 64M elements
    const int n4    = total / 4;                // vec4 work-items

    // 1/log2(a), a = 1 - dt/tau = 0.8  (negative constant, folded on host)
    const float inv_log2_a = 1.0f / log2f(1.0f - LIF_DT / LIF_TAU);

    if (n4 > 0) {
        int blocks = (n4 + 255) / 256;          // 62,500 blocks of 8 wave32s
        lif_v4_kernel<<<blocks, 256, 0, stream>>>((const v4f*)I, (v4f*)out,
                                                  n4, inv_log2_a);
    }
    const int done = n4 * 4;
    const int rem  = total - done;
    if (rem > 0) {
        int blocks = (rem + 255) / 256;
        lif_tail_kernel<<<blocks, 256, 0, stream>>>(I, out, done, total, inv_log2_a);
    }
}